// ApplyKernel_52862457479226
// MI455X (gfx1250) — compile-verified
//
#include <hip/hip_runtime.h>

// Z, N, CIN, COUT, HID = 4, 256, 32, 32, 64
#define Zc    4
#define Nc    256
#define CINc  32
#define COUTc 32
#define HIDc  64

typedef float v2f __attribute__((ext_vector_type(2)));
typedef float v8f __attribute__((ext_vector_type(8)));

// ---------------------------------------------------------------------------
// Phase 1: M[z,b,h,i] = sum_j W2[h, i*32+j] * f[z,b,j]
//          c[z,b,i]   = sum_j b2[i*32+j]   * f[z,b,j]
// One block handles 8 'b' points; W2 rows staged through LDS (padded stride 33
// so the 32 lanes reading sW2[i*33+j] hit 32 distinct banks).
// ---------------------------------------------------------------------------
__global__ __launch_bounds__(256) void hyperkernel_phase1(
    const float* __restrict__ feat, const float* __restrict__ W2,
    const float* __restrict__ b2, float* __restrict__ Mg,
    float* __restrict__ cg) {
  __shared__ float sW2[32 * 33];
  __shared__ float sF[8 * 32];
  const int z  = blockIdx.x >> 5;          // 32 blocks per z
  const int b0 = (blockIdx.x & 31) * 8;    // 8 b's per block
  const int t  = threadIdx.x;

  // features for the 8 points: sF[bo*32 + j]
  sF[t] = feat[(z * Nc + b0 + (t >> 5)) * CINc + (t & 31)];

  const int i  = t & 31;   // output channel i (COUT)
  const int bo = t >> 5;   // which of the 8 b's

  for (int h = 0; h <= HIDc; ++h) {
    const float* src = (h < HIDc) ? (W2 + (size_t)h * (COUTc * CINc)) : b2;
    __syncthreads();  // WAR on sW2 from previous iter; also covers initial sF
    // stage one 1024-float row, scattered into padded [i][j] layout
    #pragma unroll
    for (int e = 0; e < 4; ++e) {
      int idx = t * 4 + e;
      sW2[(idx >> 5) * 33 + (idx & 31)] = src[idx];
    }
    __syncthreads();
    float acc = 0.f;
    #pragma unroll
    for (int j = 0; j < CINc; ++j)
      acc += sW2[i * 33 + j] * sF[bo * 32 + j];  // sF read is a broadcast
    const int b = b0 + bo;
    if (h < HIDc)
      Mg[((size_t)(z * Nc + b)) * (HIDc * COUTc) + h * COUTc + i] = acc;
    else
      cg[(size_t)(z * Nc + b) * COUTc + i] = acc;
  }
}

// ---------------------------------------------------------------------------
// Phase 2: one block per (z,b).  out[z, a, b, i] = relu((g_b-g_a)W1 + b1) @ M
// + c, computed as two 128-row passes of a [128x64]@[64x32] FP32 WMMA GEMM.
//
// V_WMMA_F32_16X16X4_F32 fragment layout (wave32):
//   A (16x4):  lane = (half,lr): M = lr, v = A[lr][2*half + {0,1}]
//   B (4x16):  N = lr,           v = B[2*half + {0,1}][lr]
//   C (16x16): VGPR r -> M = r + 8*half, N = lr
//
// Fragment feeds are single aligned 64-bit LDS loads:
//   sH  stride 68:   A pair contiguous; halves hit disjoint bank sets.
//   sMp stride 96:   M repacked as [kpair][col*2 + (k&1)] so the B k-pair is
//                    contiguous per column; half offset 96 mod 64 = 32 ->
//                    disjoint 32-bank windows for the two lane halves.
// One A fragment is reused by both i-tile accumulators (2 WMMAs per A load).
// ---------------------------------------------------------------------------
__global__ __launch_bounds__(256) void hyperkernel_phase2(
    const float* __restrict__ geom, const float* __restrict__ W1,
    const float* __restrict__ b1, const float* __restrict__ Mg,
    const float* __restrict__ cg, float* __restrict__ out) {
  __shared__ __align__(16) float sH[128 * 68];   // 34.8 KB
  __shared__ __align__(16) float sMp[32 * 96];   // 12.3 KB, paired-k layout
  __shared__ float sW1[3 * 64];
  __shared__ float sb1[64];
  __shared__ float sc[32];

  const int z = blockIdx.x >> 8;
  const int b = blockIdx.x & 255;
  const int t = threadIdx.x;

  if (t < 192) sW1[t] = W1[t];
  if (t >= 192) sb1[t - 192] = b1[t - 192];
  if (t < 32) sc[t] = cg[(size_t)(z * Nc + b) * COUTc + t];
  for (int e = t; e < HIDc * COUTc; e += 256) {
    const int h = e >> 5, i = e & 31;
    sMp[(h >> 1) * 96 + i * 2 + (h & 1)] =
        Mg[((size_t)(z * Nc + b)) * (HIDc * COUTc) + e];
  }

  const float gbx = geom[(z * Nc + b) * 3 + 0];
  const float gby = geom[(z * Nc + b) * 3 + 1];
  const float gbz = geom[(z * Nc + b) * 3 + 2];

  const int lane = t & 31, wave = t >> 5;
  const int half = lane >> 4, lr = lane & 15;

  for (int pass = 0; pass < 2; ++pass) {
    const int abase = pass * 128;
    __syncthreads();  // pass0: shared loads done; pass1: step-C reads done

    // ---- step B: H rows for this pass (2 threads per row, 32 hh each) ----
    {
      const int al  = t >> 1;
      const int hh0 = (t & 1) * 32;
      const int a   = abase + al;
      const float d0 = gbx - geom[(z * Nc + a) * 3 + 0];
      const float d1 = gby - geom[(z * Nc + a) * 3 + 1];
      const float d2 = gbz - geom[(z * Nc + a) * 3 + 2];
      #pragma unroll
      for (int hh = hh0; hh < hh0 + 32; hh += 4) {
        float4 v;
        v.x = fmaxf(d0 * sW1[hh + 0] + d1 * sW1[64 + hh + 0] +
                        d2 * sW1[128 + hh + 0] + sb1[hh + 0], 0.f);
        v.y = fmaxf(d0 * sW1[hh + 1] + d1 * sW1[64 + hh + 1] +
                        d2 * sW1[128 + hh + 1] + sb1[hh + 1], 0.f);
        v.z = fmaxf(d0 * sW1[hh + 2] + d1 * sW1[64 + hh + 2] +
                        d2 * sW1[128 + hh + 2] + sb1[hh + 2], 0.f);
        v.w = fmaxf(d0 * sW1[hh + 3] + d1 * sW1[64 + hh + 3] +
                        d2 * sW1[128 + hh + 3] + sb1[hh + 3], 0.f);
        *reinterpret_cast<float4*>(&sH[al * 68 + hh]) = v;
      }
    }
    __syncthreads();

    // ---- step C: wave owns a-tile `wave`; two i-tile accumulators share A.
    const int rowl = wave * 16 + lr;  // local A row for this wave's tile
    v8f acc0, acc1;
    {
      const float c0 = sc[lr], c1 = sc[16 + lr];
      #pragma unroll
      for (int r = 0; r < 8; ++r) { acc0[r] = c0; acc1[r] = c1; }
    }

    #pragma unroll
    for (int k0 = 0; k0 < HIDc; k0 += 4) {
      const int kb = k0 + half * 2;   // even
      const int kp = kb >> 1;
      const v2f Av = *reinterpret_cast<const v2f*>(&sH[rowl * 68 + kb]);
      const v2f B0 = *reinterpret_cast<const v2f*>(&sMp[kp * 96 + lr * 2]);
      const v2f B1 = *reinterpret_cast<const v2f*>(&sMp[kp * 96 + 32 + lr * 2]);
      acc0 = __builtin_amdgcn_wmma_f32_16x16x4_f32(
          false, Av, false, B0, (short)0, acc0, false, false);
      acc1 = __builtin_amdgcn_wmma_f32_16x16x4_f32(
          false, Av, false, B1, (short)0, acc1, false, false);
    }

    const int a0 = abase + wave * 16 + half * 8;
    #pragma unroll
    for (int r = 0; r < 8; ++r) {
      const int a = a0 + r;
      const size_t base = (((size_t)z * Nc + a) * Nc + b) * COUTc;
      out[base + lr]      = acc0[r];
      out[base + 16 + lr] = acc1[r];
    }
  }
}

// ---------------------------------------------------------------------------
extern "C" void kernel_launch(void* const* d_in, const int* in_sizes, int n_in,
                              void* d_out, int out_size, void* d_ws,
                              size_t ws_size, hipStream_t stream) {
  const float* feat = (const float*)d_in[0];  // [Z,N,CIN]
  const float* geom = (const float*)d_in[1];  // [Z,N,3]
  const float* W1   = (const float*)d_in[2];  // [3,HID]
  const float* b1   = (const float*)d_in[3];  // [HID]
  const float* W2   = (const float*)d_in[4];  // [HID, COUT*CIN]
  const float* b2   = (const float*)d_in[5];  // [COUT*CIN]
  float* out = (float*)d_out;                 // [Z,N,N,COUT]

  float* Mg = (float*)d_ws;                              // Z*N*HID*COUT f32 (8 MB)
  float* cg = Mg + (size_t)Zc * Nc * HIDc * COUTc;       // Z*N*COUT f32

  hyperkernel_phase1<<<dim3(Zc * Nc / 8), dim3(256), 0, stream>>>(
      feat, W2, b2, Mg, cg);
  hyperkernel_phase2<<<dim3(Zc * Nc), dim3(256), 0, stream>>>(
      geom, W1, b1, Mg, cg, out);
}